// Loss5_54717883351221
// MI455X (gfx1250) — compile-verified
//
#include <hip/hip_runtime.h>

#define BATCH 4096
#define NCLS  50257
#define CAP   3072
#define TOPC  8

typedef float v4f __attribute__((ext_vector_type(4)));

// ---------- branchless sorted top-8 helpers (wave32) ----------

__device__ __forceinline__ void insert8(float s[TOPC], float v) {
#pragma unroll
  for (int i = 0; i < TOPC; ++i) {
    float hi = fmaxf(s[i], v);
    v = fminf(s[i], v);
    s[i] = hi;
  }
}

// merge my sorted-desc 8-list with lane^mask partner's; both sides end with
// the top-8 of the union. r[i] = max(A[i], B[i], max_{j<i} min(A[j], B[i-1-j]))
__device__ __forceinline__ void merge8_xor(float s[TOPC], int mask) {
  float b[TOPC];
#pragma unroll
  for (int j = 0; j < TOPC; ++j) b[j] = __shfl_xor(s[j], mask, 32);
  float r[TOPC];
#pragma unroll
  for (int i = 0; i < TOPC; ++i) {
    float m = fmaxf(s[i], b[i]);
#pragma unroll
    for (int j = 0; j < i; ++j) m = fmaxf(m, fminf(s[j], b[i - 1 - j]));
    r[i] = m;
  }
#pragma unroll
  for (int i = 0; i < TOPC; ++i) s[i] = r[i];
}

__device__ __forceinline__ void wave_top8(float s[TOPC]) {
  merge8_xor(s, 1);
  merge8_xor(s, 2);
  merge8_xor(s, 4);
  merge8_xor(s, 8);
  merge8_xor(s, 16);
}

__device__ __forceinline__ void push_cand(float* cand, int* cnt, float v) {
  int p = atomicAdd(cnt, 1);
  if (p < CAP) cand[p] = v;
}

// ---------- kernel 1: per-row exact (k+1)-th largest + gather ----------
// One block per row. Seeded-threshold single pass: threshold T = 8th largest
// of 256 per-thread max4 values guarantees >=8 row elements >= T, so the
// row's top-8 are all captured by appending v >= T. Exact; fallback rescans.

__global__ void __launch_bounds__(256) topk_gather_kernel(
    const float* __restrict__ x, const int* __restrict__ y,
    const int* __restrict__ kp, float* __restrict__ sa,
    float* __restrict__ sb, double* __restrict__ acc) {
  __shared__ float cand[CAP];
  __shared__ float maxes[256];
  __shared__ float shT;
  __shared__ int shCount;

  const int row = blockIdx.x;
  const int tid = threadIdx.x;
  const float* rowp = x + (size_t)row * NCLS;

  if (row == 0 && tid == 0) *acc = 0.0;           // zero fp64 accumulator once
  if (tid == 0) {
    sb[row] = rowp[y[row]];                       // s_y gather
    shCount = 0;
  }

  // peel to 16B alignment (rows are only 4B aligned: 50257 floats/row)
  const uintptr_t ab = (uintptr_t)rowp;
  const int head = (int)((4u - ((unsigned)(ab >> 2) & 3u)) & 3u);
  const int n4 = (NCLS - head) >> 2;
  const v4f* row4 = (const v4f*)(rowp + head);
  const int tail0 = head + (n4 << 2);

  // ---- seed threshold ----
  v4f v0 = __builtin_nontemporal_load(row4 + tid);
  maxes[tid] = fmaxf(fmaxf(v0.x, v0.y), fmaxf(v0.z, v0.w));
  __syncthreads();
  if (tid < 32) {
    float s[TOPC];
#pragma unroll
    for (int i = 0; i < TOPC; ++i) s[i] = -__builtin_inff();
#pragma unroll
    for (int r = 0; r < 8; ++r) insert8(s, maxes[tid + 32 * r]);
    wave_top8(s);
    if (tid == 0) shT = s[TOPC - 1];
  }
  __syncthreads();
  const float T = shT;

  // ---- single streaming pass: append candidates >= T ----
  if (v0.x >= T) push_cand(cand, &shCount, v0.x);
  if (v0.y >= T) push_cand(cand, &shCount, v0.y);
  if (v0.z >= T) push_cand(cand, &shCount, v0.z);
  if (v0.w >= T) push_cand(cand, &shCount, v0.w);
  for (int e4 = tid + 256; e4 < n4; e4 += 256) {
    v4f v = __builtin_nontemporal_load(row4 + e4);
    float g = fmaxf(fmaxf(v.x, v.y), fmaxf(v.z, v.w));
    if (g >= T) {
      if (v.x >= T) push_cand(cand, &shCount, v.x);
      if (v.y >= T) push_cand(cand, &shCount, v.y);
      if (v.z >= T) push_cand(cand, &shCount, v.z);
      if (v.w >= T) push_cand(cand, &shCount, v.w);
    }
  }
  // head/tail scalars (<=7 total)
  if (tid < head) {
    float v = rowp[tid];
    if (v >= T) push_cand(cand, &shCount, v);
  }
  {
    int ti = tail0 + tid;
    if (ti < NCLS) {
      float v = rowp[ti];
      if (v >= T) push_cand(cand, &shCount, v);
    }
  }
  __syncthreads();

  const int cnt = shCount;
  int kk = kp[0];
  if (kk < 0) kk = 0;
  if (kk > TOPC - 1) kk = TOPC - 1;

  // ---- final selection by one wave ----
  if (tid < 32) {
    float s[TOPC];
#pragma unroll
    for (int i = 0; i < TOPC; ++i) s[i] = -__builtin_inff();
    if (cnt <= CAP) {
      for (int i = tid; i < cnt; i += 32) insert8(s, cand[i]);
    } else {
      // overflow fallback (never for Gaussian input): exact full rescan
      for (int i = tid; i < NCLS; i += 32) insert8(s, rowp[i]);
    }
    wave_top8(s);
    if (tid == 0) {
      float r = s[TOPC - 1];
#pragma unroll
      for (int i = 0; i < TOPC; ++i)
        if (i == kk) r = s[i];
      sa[row] = r;
    }
  }
}

// ---------- kernel 2: mean over [B,B] of relu(1 + a[j] - b[i]) ----------
// grid (16,16): x = j-tile, y = i-tile. a-tile staged via CDNA5 async-to-LDS.

__global__ void __launch_bounds__(256) pair_loss_kernel(
    const float* __restrict__ sa, const float* __restrict__ sb,
    double* __restrict__ acc) {
  __shared__ float aj[256];
  const int tid = threadIdx.x;
  const int jbase = blockIdx.x * 256;
  const int i = blockIdx.y * 256 + tid;

  {
    // low 32 bits of a generic LDS pointer == wave-relative LDS byte offset
    unsigned lds = (unsigned)(size_t)&aj[tid];
    unsigned voff = (unsigned)(tid * 4);
    const float* base = sa + jbase;
    asm volatile("global_load_async_to_lds_b32 %0, %1, %2"
                 :: "v"(lds), "v"(voff), "s"(base)
                 : "memory");
    asm volatile("s_wait_asynccnt 0" ::: "memory");
  }
  __syncthreads();

  const float bv = sb[i];
  float p = 0.0f;
#pragma unroll 8
  for (int j = 0; j < 256; ++j) p += fmaxf(1.0f + aj[j] - bv, 0.0f);

#pragma unroll
  for (int m = 16; m >= 1; m >>= 1) p += __shfl_xor(p, m, 32);
  if ((tid & 31) == 0) unsafeAtomicAdd(acc, (double)p);
}

// ---------- kernel 3: finalize mean ----------

__global__ void finalize_kernel(const double* __restrict__ acc,
                                float* __restrict__ out) {
  out[0] = (float)(acc[0] * (1.0 / ((double)BATCH * (double)BATCH)));
}

extern "C" void kernel_launch(void* const* d_in, const int* in_sizes, int n_in,
                              void* d_out, int out_size, void* d_ws, size_t ws_size,
                              hipStream_t stream) {
  const float* x = (const float*)d_in[0];
  const int* y = (const int*)d_in[1];
  const int* kp = (const int*)d_in[2];
  float* out = (float*)d_out;

  double* acc = (double*)d_ws;                   // 8B fp64 accumulator
  float* sa = (float*)((char*)d_ws + 16);        // s_topk[4096]
  float* sb = sa + BATCH;                        // s_y[4096]

  hipLaunchKernelGGL(topk_gather_kernel, dim3(BATCH), dim3(256), 0, stream,
                     x, y, kp, sa, sb, acc);
  hipLaunchKernelGGL(pair_loss_kernel, dim3(16, 16), dim3(256), 0, stream,
                     sa, sb, acc);
  hipLaunchKernelGGL(finalize_kernel, dim3(1), dim3(1), 0, stream, acc, out);

  (void)in_sizes; (void)n_in; (void)out_size; (void)ws_size;
}